// GAT_26414048870740
// MI455X (gfx1250) — compile-verified
//
#include <hip/hip_runtime.h>
#include <hip/hip_bf16.h>

// ---------------- problem constants ----------------
#define NN      50000
#define EE      800000
#define ET      (EE + NN)        // edges incl. self loops = 850000
#define IN_FEAT 128
#define HID     64
#define HEADS   4
#define F1      (HEADS * HID)    // 256
#define CLS     40
#define CLS_PAD 48               // 3 x 16 col tiles
#define NEG_SLOPE 0.2f

typedef __attribute__((ext_vector_type(2))) float v2f;
typedef __attribute__((ext_vector_type(8))) float v8f;

// ---- order-preserving float <-> uint encoding for atomicMax ----
__device__ __forceinline__ unsigned fenc(float f) {
    unsigned u = __float_as_uint(f);
    return (u & 0x80000000u) ? ~u : (u | 0x80000000u);
}
__device__ __forceinline__ float fdec(unsigned u) {
    unsigned v = (u & 0x80000000u) ? (u & 0x7fffffffu) : ~u;
    return __uint_as_float(v);
}
__device__ __forceinline__ float lrelu(float v) { return v > 0.f ? v : NEG_SLOPE * v; }

__device__ __forceinline__ void edge_sd(int e, const int* src, const int* dst, int& s, int& d) {
    if (e < EE) { s = src[e]; d = dst[e]; } else { s = d = e - EE; }
}

// ---------------- utility: zero fill ----------------
__global__ void zero_u32_kernel(unsigned* p, long long n) {
    long long i = (long long)blockIdx.x * blockDim.x + threadIdx.x;
    long long stride = (long long)gridDim.x * blockDim.x;
    for (; i < n; i += stride) p[i] = 0u;
}

// ---------------- GEMM1: fea1[N,256] = X[N,128] @ W1[128,256] + b1 ----------------
// grid 3125 blocks (16-row bands), 512 threads = 16 waves, wave w -> col tile w
__global__ void gemm1_wmma_kernel(const float* __restrict__ X,
                                  const float* __restrict__ W,
                                  const float* __restrict__ b,
                                  float* __restrict__ fea) {
    __shared__ float As[16][132];                 // padded: bank-conflict free
    const int m0 = blockIdx.x * 16;
    for (int i = threadIdx.x; i < 16 * IN_FEAT; i += blockDim.x) {
        int r = i >> 7, c = i & 127;
        As[r][c] = X[(size_t)(m0 + r) * IN_FEAT + c];
    }
    __syncthreads();

    const int wave = threadIdx.x >> 5;
    const int lane = threadIdx.x & 31;
    const int col  = wave * 16 + (lane & 15);     // 0..255
    const int mrow = lane & 15;
    const int kh   = (lane >> 4) * 2;             // 0 or 2

    v8f acc = {};
#pragma unroll
    for (int k0 = 0; k0 < IN_FEAT; k0 += 4) {
        v2f a, bb;
        a.x  = As[mrow][k0 + kh];
        a.y  = As[mrow][k0 + kh + 1];
        bb.x = W[(size_t)(k0 + kh) * F1 + col];
        bb.y = W[(size_t)(k0 + kh + 1) * F1 + col];
        acc = __builtin_amdgcn_wmma_f32_16x16x4_f32(false, a, false, bb, 0, acc, false, false);
    }
    const int rbase = (lane >> 4) * 8;
    const float bias = b[col];
#pragma unroll
    for (int r = 0; r < 8; ++r) {
        int gr = m0 + rbase + r;                  // N is an exact multiple of 16
        fea[(size_t)gr * F1 + col] = acc[r] + bias;
    }
}

// ---------------- GEMM2: fea2[N,40] = H[N,256] @ W2[256,40] + b2 ----------------
// grid 3125 blocks, 96 threads = 3 waves (col tiles 0..2, cols padded to 48)
__global__ void gemm2_wmma_kernel(const float* __restrict__ Hin,
                                  const float* __restrict__ W,
                                  const float* __restrict__ b,
                                  float* __restrict__ fea) {
    __shared__ float As[16][260];
    const int m0 = blockIdx.x * 16;
    for (int i = threadIdx.x; i < 16 * F1; i += blockDim.x) {
        int r = i >> 8, c = i & 255;
        As[r][c] = Hin[(size_t)(m0 + r) * F1 + c];
    }
    __syncthreads();

    const int wave = threadIdx.x >> 5;
    const int lane = threadIdx.x & 31;
    const int col  = wave * 16 + (lane & 15);     // 0..47 (only <40 valid)
    const int mrow = lane & 15;
    const int kh   = (lane >> 4) * 2;
    const bool ok  = (col < CLS);

    v8f acc = {};
#pragma unroll
    for (int k0 = 0; k0 < F1; k0 += 4) {
        v2f a, bb;
        a.x  = As[mrow][k0 + kh];
        a.y  = As[mrow][k0 + kh + 1];
        bb.x = ok ? W[(size_t)(k0 + kh) * CLS + col] : 0.f;
        bb.y = ok ? W[(size_t)(k0 + kh + 1) * CLS + col] : 0.f;
        acc = __builtin_amdgcn_wmma_f32_16x16x4_f32(false, a, false, bb, 0, acc, false, false);
    }
    if (ok) {
        const int rbase = (lane >> 4) * 8;
        const float bias = b[col];
#pragma unroll
        for (int r = 0; r < 8; ++r) {
            int gr = m0 + rbase + r;
            fea[(size_t)gr * CLS + col] = acc[r] + bias;
        }
    }
}

// ---------------- per-node attention scores ----------------
// si[n,h] = dot(fea[n,h,:], att[h, 0:C]);  sj[n,h] = dot(fea[n,h,:], att[h, C:2C])
__global__ void scores_kernel(const float* __restrict__ fea, const float* __restrict__ att,
                              float* __restrict__ si, float* __restrict__ sj,
                              int H, int C) {
    long long idx = (long long)blockIdx.x * blockDim.x + threadIdx.x;
    if (idx >= (long long)NN * H) return;
    int n = (int)(idx / H);
    int h = (int)(idx - (long long)n * H);
    const float* f  = fea + (size_t)n * H * C + (size_t)h * C;
    const float* ai = att + (size_t)h * 2 * C;
    const float* aj = ai + C;
    float vi = 0.f, vj = 0.f;
    for (int c = 0; c < C; ++c) {
        float x = f[c];
        vi += x * ai[c];
        vj += x * aj[c];
    }
    si[idx] = vi;
    sj[idx] = vj;
}

// ---------------- edge passes: leaky-relu + segment max ----------------
__global__ void edge_max_kernel(const int* __restrict__ src, const int* __restrict__ dst,
                                const float* __restrict__ si, const float* __restrict__ sj,
                                float* __restrict__ alpha, unsigned* __restrict__ amax, int H) {
    long long idx = (long long)blockIdx.x * blockDim.x + threadIdx.x;
    if (idx >= (long long)ET * H) return;
    int e = (int)(idx / H);
    int h = (int)(idx - (long long)e * H);
    int s, d; edge_sd(e, src, dst, s, d);
    float v = lrelu(si[(size_t)d * H + h] + sj[(size_t)s * H + h]);
    alpha[idx] = v;
    atomicMax(&amax[(size_t)d * H + h], fenc(v));
}

// ---------------- edge passes: exp + segment sum ----------------
__global__ void edge_exp_kernel(const int* __restrict__ src, const int* __restrict__ dst,
                                float* __restrict__ alpha, const unsigned* __restrict__ amax,
                                float* __restrict__ asum, int H) {
    long long idx = (long long)blockIdx.x * blockDim.x + threadIdx.x;
    if (idx >= (long long)ET * H) return;
    int e = (int)(idx / H);
    int h = (int)(idx - (long long)e * H);
    int s, d; edge_sd(e, src, dst, s, d);
    float ex = expf(alpha[idx] - fdec(amax[(size_t)d * H + h]));
    alpha[idx] = ex;
    atomicAdd(&asum[(size_t)d * H + h], ex);
}

// ---------------- edge passes: normalize ----------------
__global__ void edge_norm_kernel(const int* __restrict__ src, const int* __restrict__ dst,
                                 float* __restrict__ alpha, const float* __restrict__ asum, int H) {
    long long idx = (long long)blockIdx.x * blockDim.x + threadIdx.x;
    if (idx >= (long long)ET * H) return;
    int e = (int)(idx / H);
    int h = (int)(idx - (long long)e * H);
    int s, d; edge_sd(e, src, dst, s, d);
    alpha[idx] = alpha[idx] / (asum[(size_t)d * H + h] + 1e-16f);
}

// ---------------- scatter: out[dst, ch] += fea[src, ch] * alpha[e, h(ch)] ----------------
// layer 1: F=256, H=4 (h = ch>>6);  one thread per (edge, channel)
__global__ void aggregate1_kernel(const int* __restrict__ src, const int* __restrict__ dst,
                                  const float* __restrict__ fea, const float* __restrict__ alpha,
                                  float* __restrict__ out) {
    long long idx = (long long)blockIdx.x * blockDim.x + threadIdx.x;
    if (idx >= (long long)ET * F1) return;
    int e  = (int)(idx >> 8);
    int ch = (int)(idx & 255);
    int s, d; edge_sd(e, src, dst, s, d);
    int h = ch >> 6;
    float a = alpha[(size_t)e * HEADS + h];
    atomicAdd(&out[(size_t)d * F1 + ch], fea[(size_t)s * F1 + ch] * a);
}

// layer 2: F=40, H=1
__global__ void aggregate2_kernel(const int* __restrict__ src, const int* __restrict__ dst,
                                  const float* __restrict__ fea, const float* __restrict__ alpha,
                                  float* __restrict__ out) {
    long long idx = (long long)blockIdx.x * blockDim.x + threadIdx.x;
    if (idx >= (long long)ET * CLS) return;
    int e  = (int)(idx / CLS);
    int ch = (int)(idx - (long long)e * CLS);
    int s, d; edge_sd(e, src, dst, s, d);
    atomicAdd(&out[(size_t)d * CLS + ch], fea[(size_t)s * CLS + ch] * alpha[e]);
}

// ---------------- h = relu(agg + bias1) in place ----------------
__global__ void relu_bias_kernel(float* __restrict__ buf, const float* __restrict__ bias) {
    long long idx = (long long)blockIdx.x * blockDim.x + threadIdx.x;
    if (idx >= (long long)NN * F1) return;
    float v = buf[idx] + bias[(int)(idx & 255)];
    buf[idx] = v > 0.f ? v : 0.f;
}

// ---------------- final: y = softmax(out2 + bias2, axis=1) ----------------
__global__ void softmax_out_kernel(const float* __restrict__ out2, const float* __restrict__ bias,
                                   float* __restrict__ y) {
    int n = blockIdx.x * blockDim.x + threadIdx.x;
    if (n >= NN) return;
    float v[CLS];
    float m = -3.4e38f;
    for (int c = 0; c < CLS; ++c) {
        v[c] = out2[(size_t)n * CLS + c] + bias[c];
        m = fmaxf(m, v[c]);
    }
    float s = 0.f;
    for (int c = 0; c < CLS; ++c) { v[c] = expf(v[c] - m); s += v[c]; }
    float inv = 1.f / s;
    for (int c = 0; c < CLS; ++c) y[(size_t)n * CLS + c] = v[c] * inv;
}

// =====================================================================
extern "C" void kernel_launch(void* const* d_in, const int* in_sizes, int n_in,
                              void* d_out, int out_size, void* d_ws, size_t ws_size,
                              hipStream_t stream) {
    const float* x      = (const float*)d_in[0];
    const int*   ei     = (const int*)  d_in[1];   // [2, E] flat
    const float* lin1_w = (const float*)d_in[2];
    const float* lin1_b = (const float*)d_in[3];
    const float* att1   = (const float*)d_in[4];
    const float* bias1  = (const float*)d_in[5];
    const float* lin2_w = (const float*)d_in[6];
    const float* lin2_b = (const float*)d_in[7];
    const float* att2   = (const float*)d_in[8];
    const float* bias2  = (const float*)d_in[9];
    float* y = (float*)d_out;

    const int* src = ei;
    const int* dst = ei + EE;

    // ---- workspace layout ----
    size_t off = 0;
    auto alloc = [&](size_t bytes) {
        size_t p = off;
        off += (bytes + 255) & ~(size_t)255;
        return p;
    };
    char* ws = (char*)d_ws;
    float*    fea1   = (float*)   (ws + alloc((size_t)NN * F1 * 4));
    float*    hbuf   = (float*)   (ws + alloc((size_t)NN * F1 * 4));   // agg1 -> relu -> layer-2 input
    float*    s1i    = (float*)   (ws + alloc((size_t)NN * HEADS * 4));
    float*    s1j    = (float*)   (ws + alloc((size_t)NN * HEADS * 4));
    unsigned* amax1  = (unsigned*)(ws + alloc((size_t)NN * HEADS * 4));
    float*    sum1   = (float*)   (ws + alloc((size_t)NN * HEADS * 4));
    float*    alpha1 = (float*)   (ws + alloc((size_t)ET * HEADS * 4));
    float*    fea2   = (float*)   (ws + alloc((size_t)NN * CLS * 4));
    float*    out2   = (float*)   (ws + alloc((size_t)NN * CLS * 4));
    float*    s2i    = (float*)   (ws + alloc((size_t)NN * 4));
    float*    s2j    = (float*)   (ws + alloc((size_t)NN * 4));
    unsigned* amax2  = (unsigned*)(ws + alloc((size_t)NN * 4));
    float*    sum2   = (float*)   (ws + alloc((size_t)NN * 4));
    float*    alpha2 = (float*)   (ws + alloc((size_t)ET * 4));
    (void)ws_size; (void)in_sizes; (void)n_in; (void)out_size;

    const int TB = 256;
    auto blocks = [](long long n, int tb) { return (unsigned)((n + tb - 1) / tb); };

    // ---------- layer 1 ----------
    zero_u32_kernel<<<2048, TB, 0, stream>>>((unsigned*)amax1, (long long)NN * HEADS);
    zero_u32_kernel<<<2048, TB, 0, stream>>>((unsigned*)sum1,  (long long)NN * HEADS);
    zero_u32_kernel<<<8192, TB, 0, stream>>>((unsigned*)hbuf,  (long long)NN * F1);

    gemm1_wmma_kernel<<<NN / 16, 512, 0, stream>>>(x, lin1_w, lin1_b, fea1);
    scores_kernel<<<blocks((long long)NN * HEADS, TB), TB, 0, stream>>>(fea1, att1, s1i, s1j, HEADS, HID);

    edge_max_kernel <<<blocks((long long)ET * HEADS, TB), TB, 0, stream>>>(src, dst, s1i, s1j, alpha1, amax1, HEADS);
    edge_exp_kernel <<<blocks((long long)ET * HEADS, TB), TB, 0, stream>>>(src, dst, alpha1, amax1, sum1, HEADS);
    edge_norm_kernel<<<blocks((long long)ET * HEADS, TB), TB, 0, stream>>>(src, dst, alpha1, sum1, HEADS);
    aggregate1_kernel<<<blocks((long long)ET * F1, TB), TB, 0, stream>>>(src, dst, fea1, alpha1, hbuf);
    relu_bias_kernel<<<blocks((long long)NN * F1, TB), TB, 0, stream>>>(hbuf, bias1);

    // ---------- layer 2 ----------
    zero_u32_kernel<<<2048, TB, 0, stream>>>((unsigned*)amax2, (long long)NN);
    zero_u32_kernel<<<2048, TB, 0, stream>>>((unsigned*)sum2,  (long long)NN);
    zero_u32_kernel<<<4096, TB, 0, stream>>>((unsigned*)out2,  (long long)NN * CLS);

    gemm2_wmma_kernel<<<NN / 16, 96, 0, stream>>>(hbuf, lin2_w, lin2_b, fea2);
    scores_kernel<<<blocks((long long)NN, TB), TB, 0, stream>>>(fea2, att2, s2i, s2j, 1, CLS);

    edge_max_kernel <<<blocks((long long)ET, TB), TB, 0, stream>>>(src, dst, s2i, s2j, alpha2, amax2, 1);
    edge_exp_kernel <<<blocks((long long)ET, TB), TB, 0, stream>>>(src, dst, alpha2, amax2, sum2, 1);
    edge_norm_kernel<<<blocks((long long)ET, TB), TB, 0, stream>>>(src, dst, alpha2, sum2, 1);
    aggregate2_kernel<<<blocks((long long)ET * CLS, TB), TB, 0, stream>>>(src, dst, fea2, alpha2, out2);

    softmax_out_kernel<<<blocks((long long)NN, TB), TB, 0, stream>>>(out2, bias2, y);
}